// SingleStepDecoderWrapper_31894427140834
// MI455X (gfx1250) — compile-verified
//
#include <hip/hip_runtime.h>
#include <hip/hip_bf16.h>
#include <cstddef>

#define D_MODEL  1024
#define N_HEADS  16
#define HEAD_DIM 64
#define N_LAYERS 12
#define SRC_LEN  2048
#define VOCAB    50257
#define D_INNER  2048
#define D_STATE  16
#define D_CONV   4
#define DT_RANK  64
#define D_FF     4096

typedef float v2f __attribute__((ext_vector_type(2)));
typedef float v4f __attribute__((ext_vector_type(4)));
typedef float v8f __attribute__((ext_vector_type(8)));

// ---------------- helpers ----------------
__device__ __forceinline__ float wave_sum(float v) {
#pragma unroll
  for (int o = 16; o > 0; o >>= 1) v += __shfl_xor(v, o, 32);
  return v;
}
__device__ __forceinline__ float wave_max(float v) {
#pragma unroll
  for (int o = 16; o > 0; o >>= 1) v = fmaxf(v, __shfl_xor(v, o, 32));
  return v;
}
__device__ __forceinline__ float silu_f(float v)     { return v / (1.0f + expf(-v)); }
__device__ __forceinline__ float softplus_f(float v) { return v > 20.0f ? v : log1pf(expf(v)); }
__device__ __forceinline__ float gelu_f(float v) {   // tanh approximation (jax.nn.gelu default)
  const float c = 0.7978845608028654f;
  return 0.5f * v * (1.0f + tanhf(c * (v + 0.044715f * v * v * v)));
}

// ---------------- embedding ----------------
__global__ void k_embed(const int* __restrict__ tok, const float* __restrict__ emb,
                        const float* __restrict__ scale, float* __restrict__ x) {
  int i = blockIdx.x * blockDim.x + threadIdx.x;
  if (i < D_MODEL) x[i] = emb[(size_t)tok[0] * D_MODEL + i] * scale[0];
}

// ---------------- layernorm over 1024 elements, one block of 1024 threads ----------------
__global__ void __launch_bounds__(1024) k_layernorm(const float* __restrict__ x,
                                                    const float* __restrict__ w,
                                                    const float* __restrict__ b,
                                                    float* __restrict__ out) {
  __shared__ float sm[32], sv[32];
  int t = threadIdx.x, lane = t & 31, wid = t >> 5;
  float v  = x[t];
  float s1 = wave_sum(v);
  float s2 = wave_sum(v * v);
  if (lane == 0) { sm[wid] = s1; sv[wid] = s2; }
  __syncthreads();
  if (t < 32) {
    float a = wave_sum(sm[lane]);
    float c = wave_sum(sv[lane]);
    if (lane == 0) { sm[0] = a * (1.0f / D_MODEL); sv[0] = c * (1.0f / D_MODEL); }
  }
  __syncthreads();
  float mean = sm[0];
  float var  = sv[0] - mean * mean;
  out[t] = (v - mean) * rsqrtf(var + 1e-5f) * w[t] + b[t];
}

// ---------------- generic GEMV: out[row] (op)= W[row,:] . x + bias ----------------
// MODE 0: out = acc(+bias)   MODE 1: out += acc(+bias)  (residual)
// MODE 2: out = gelu(acc+bias)   MODE 3: out = softplus(acc+bias)
template <int MODE>
__global__ void __launch_bounds__(256) k_gemv(const float* __restrict__ W,
                                              const float* __restrict__ x,
                                              const float* __restrict__ bias,
                                              float* __restrict__ out,
                                              int rows, int cols) {
  int gt = blockIdx.x * blockDim.x + threadIdx.x;
  int row = gt >> 5, lane = gt & 31;
  if (row >= rows) return;
  const float* wr = W + (size_t)row * cols;
  float acc = 0.f;
  for (int k = lane * 4; k < cols; k += 128) {
    v4f wv = __builtin_nontemporal_load((const v4f*)(wr + k));   // weights: streamed once
    float4 xv = *(const float4*)(x + k);                          // activations: cache-friendly
    acc += wv.x * xv.x + wv.y * xv.y + wv.z * xv.z + wv.w * xv.w;
  }
  acc = wave_sum(acc);
  if (lane == 0) {
    if (bias) acc += bias[row];
    if (MODE == 1)      out[row] += acc;
    else if (MODE == 2) out[row]  = gelu_f(acc);
    else if (MODE == 3) out[row]  = softplus_f(acc);
    else                out[row]  = acc;
  }
}

// ---------------- mamba depthwise conv (width 4) + silu; emits conv_out state ----------------
__global__ void __launch_bounds__(256) k_mamba_conv(const float* __restrict__ xz,
                                                    const float* __restrict__ conv_state,
                                                    const float* __restrict__ conv_w,
                                                    const float* __restrict__ conv_b,
                                                    float* __restrict__ x_conv,
                                                    float* __restrict__ conv_out) {
  int i = blockIdx.x * blockDim.x + threadIdx.x;
  if (i >= D_INNER) return;
  float c0 = conv_state[i * 3 + 0], c1 = conv_state[i * 3 + 1], c2 = conv_state[i * 3 + 2];
  float xp = xz[i];  // x_part
  float s = c0 * conv_w[i * 4 + 0] + c1 * conv_w[i * 4 + 1] +
            c2 * conv_w[i * 4 + 2] + xp * conv_w[i * 4 + 3] + conv_b[i];
  x_conv[i] = silu_f(s);
  conv_out[i * 3 + 0] = c1;
  conv_out[i * 3 + 1] = c2;
  conv_out[i * 3 + 2] = xp;
}

// ---------------- selective-scan single step; emits h_out state and gated y ----------------
__global__ void __launch_bounds__(256) k_ssm(const float* __restrict__ dt,
                                             const float* __restrict__ x_dbl,
                                             const float* __restrict__ A_log,
                                             const float* __restrict__ h_in,
                                             const float* __restrict__ Dp,
                                             const float* __restrict__ x_conv,
                                             const float* __restrict__ xz,
                                             float* __restrict__ h_out,
                                             float* __restrict__ y) {
  int i = blockIdx.x * blockDim.x + threadIdx.x;
  if (i >= D_INNER) return;
  float dti = dt[i], xc = x_conv[i];
  float acc = 0.f;
#pragma unroll
  for (int s = 0; s < D_STATE; ++s) {
    float A  = -expf(A_log[i * D_STATE + s]);
    float dA = expf(A * dti);
    float hn = dA * h_in[i * D_STATE + s] + dti * x_dbl[DT_RANK + s] * xc;
    h_out[i * D_STATE + s] = hn;
    acc += hn * x_dbl[DT_RANK + D_STATE + s];   // C
  }
  float yv = acc + Dp[i] * xc;
  y[i] = yv * silu_f(xz[D_INNER + i]);          // gate with z
}

// ---------------- attention: scores[h,k] = (q_h . K[h,k,:]) * scale ----------------
__global__ void __launch_bounds__(256) k_attn_scores(const float* __restrict__ q,
                                                     const float* __restrict__ K,
                                                     float* __restrict__ scores) {
  int h   = blockIdx.x;
  int key = blockIdx.y * 8 + (threadIdx.x >> 5);
  int lane = threadIdx.x & 31;
  const float* kr = K + ((size_t)h * SRC_LEN + key) * HEAD_DIM;
  v2f kv = __builtin_nontemporal_load((const v2f*)(kr + lane * 2));  // K streamed once
  float2 qv = *(const float2*)(q + h * HEAD_DIM + lane * 2);
  float acc = wave_sum(kv.x * qv.x + kv.y * qv.y);
  if (lane == 0) scores[h * SRC_LEN + key] = acc * 0.125f;  // 1/sqrt(64)
}

// ---------------- softmax over 2048 per head, block = 1024 threads ----------------
__global__ void __launch_bounds__(1024) k_softmax(float* __restrict__ scores) {
  __shared__ float sh[32];
  __shared__ float bcast;
  int h = blockIdx.x, t = threadIdx.x, lane = t & 31, wid = t >> 5;
  float* s = scores + (size_t)h * SRC_LEN;
  float a = s[t], b = s[t + 1024];
  float m = wave_max(fmaxf(a, b));
  if (lane == 0) sh[wid] = m;
  __syncthreads();
  if (t < 32) { float mm = wave_max(sh[lane]); if (lane == 0) bcast = mm; }
  __syncthreads();
  float M = bcast;
  float e0 = expf(a - M), e1 = expf(b - M);
  __syncthreads();
  float ssum = wave_sum(e0 + e1);
  if (lane == 0) sh[wid] = ssum;
  __syncthreads();
  if (t < 32) { float tt = wave_sum(sh[lane]); if (lane == 0) bcast = tt; }
  __syncthreads();
  float inv = 1.0f / bcast;
  s[t] = e0 * inv;
  s[t + 1024] = e1 * inv;
}

// ---------------- o[h,d] = sum_k attn[h,k] * V[h,k,d] (coalesced over d) ----------------
__global__ void __launch_bounds__(256) k_attn_av(const float* __restrict__ scores,
                                                 const float* __restrict__ V,
                                                 float* __restrict__ o) {
  __shared__ float red[256];
  int h = blockIdx.x;
  int d  = threadIdx.x & 63;
  int kk = threadIdx.x >> 6;  // 0..3
  const float* s  = scores + (size_t)h * SRC_LEN;
  const float* vb = V + (size_t)h * SRC_LEN * HEAD_DIM;
  float acc = 0.f;
  for (int k = kk; k < SRC_LEN; k += 4)
    acc += s[k] * __builtin_nontemporal_load(vb + (size_t)k * HEAD_DIM + d);  // V streamed once
  red[threadIdx.x] = acc;
  __syncthreads();
  if (kk == 0) o[h * HEAD_DIM + d] = red[d] + red[64 + d] + red[128 + d] + red[192 + d];
}

// ---------------- vocab projection via WMMA f32 16x16x4, M padded 1->16 ----------------
// Each wave produces 16 logits. B tile (4x16 of out_w^T) loaded as one b128 per lane
// directly from row-major out_w; A tile has xn only in row M=0 (lanes 0 and 16),
// built branchlessly with 0/1 mask multiplies (no exec-mask divergence in the loop).
__global__ void __launch_bounds__(256) k_logits_wmma(const float* __restrict__ W,
                                                     const float* __restrict__ xn,
                                                     const float* __restrict__ bias,
                                                     float* __restrict__ out, int rows) {
  int gt = blockIdx.x * blockDim.x + threadIdx.x;
  int wave = gt >> 5, lane = gt & 31;
  int n0 = wave * 16;
  if (n0 >= rows) return;                       // wave-uniform exit: EXEC stays all-ones
  int nrow = n0 + (lane & 15);
  if (nrow >= rows) nrow = rows - 1;            // clamp (duplicate row) instead of masking
  const float* wr = W + (size_t)nrow * D_MODEL;
  bool hi = lane >= 16;
  float m0  = (lane == 0)  ? 1.f : 0.f;         // picks (M=0, K=0/1) slots
  float m16 = (lane == 16) ? 1.f : 0.f;         // picks (M=0, K=2/3) slots
  v8f c = {0.f, 0.f, 0.f, 0.f, 0.f, 0.f, 0.f, 0.f};
#pragma unroll 8
  for (int k = 0; k < D_MODEL; k += 4) {
    float4 xv = *(const float4*)(xn + k);       // uniform broadcast load (SMEM-friendly)
    v4f wv = __builtin_nontemporal_load((const v4f*)(wr + k));  // b128/lane: W[nrow][k..k+3]
    v2f a, b;
    a.x = xv.x * m0 + xv.z * m16;               // (M=0,K=0) lane0 / (M=0,K=2) lane16, else 0
    a.y = xv.y * m0 + xv.w * m16;               // (M=0,K=1) lane0 / (M=0,K=3) lane16, else 0
    b.x = hi ? wv.z : wv.x;                     // (K=0,N) lanes0-15 / (K=2,N) lanes16-31
    b.y = hi ? wv.w : wv.y;                     // (K=1,N) lanes0-15 / (K=3,N) lanes16-31
    c = __builtin_amdgcn_wmma_f32_16x16x4_f32(false, a, false, b, (short)0, c, false, false);
  }
  // D row M=0 lives in accumulator VGPR0 on lanes 0..15
  if (lane < 16 && (n0 + lane) < rows)
    out[n0 + lane] = c[0] + bias[n0 + lane];
}

// ---------------- host-side orchestration ----------------
extern "C" void kernel_launch(void* const* d_in, const int* in_sizes, int n_in,
                              void* d_out, int out_size, void* d_ws, size_t ws_size,
                              hipStream_t stream) {
  (void)in_sizes; (void)n_in; (void)out_size; (void)ws_size;
  const int*   tok        = (const int*)  d_in[0];
  const float* all_k      = (const float*)d_in[1];
  const float* all_v      = (const float*)d_in[2];
  const float* all_h      = (const float*)d_in[3];
  const float* all_conv   = (const float*)d_in[4];
  const float* emb        = (const float*)d_in[5];
  const float* pos_scale  = (const float*)d_in[6];
  const float* fn_w       = (const float*)d_in[7];
  const float* fn_b       = (const float*)d_in[8];
  const float* out_w      = (const float*)d_in[9];
  const float* out_b      = (const float*)d_in[10];
  const float* self_nw    = (const float*)d_in[11];
  const float* self_nb    = (const float*)d_in[12];
  const float* in_proj_w  = (const float*)d_in[13];
  const float* conv_w     = (const float*)d_in[14];
  const float* conv_b     = (const float*)d_in[15];
  const float* x_proj_w   = (const float*)d_in[16];
  const float* dt_proj_w  = (const float*)d_in[17];
  const float* dt_proj_b  = (const float*)d_in[18];
  const float* A_log      = (const float*)d_in[19];
  const float* Dp         = (const float*)d_in[20];
  const float* mamba_ow   = (const float*)d_in[21];
  const float* cross_nw   = (const float*)d_in[22];
  const float* cross_nb   = (const float*)d_in[23];
  const float* wq         = (const float*)d_in[24];
  const float* bq         = (const float*)d_in[25];
  const float* attn_ow    = (const float*)d_in[26];
  const float* attn_ob    = (const float*)d_in[27];
  const float* ff_nw      = (const float*)d_in[28];
  const float* ff_nb      = (const float*)d_in[29];
  const float* ff_w1      = (const float*)d_in[30];
  const float* ff_b1      = (const float*)d_in[31];
  const float* ff_w2      = (const float*)d_in[32];
  const float* ff_b2      = (const float*)d_in[33];

  float* ws      = (float*)d_ws;
  float* x       = ws;            // 1024
  float* xn      = ws + 1024;     // 1024
  float* xz      = ws + 2048;     // 4096
  float* x_conv  = ws + 6144;     // 2048
  float* x_dbl   = ws + 8192;     // 96 (pad to 128)
  float* dt      = ws + 8320;     // 2048
  float* y       = ws + 10368;    // 2048
  float* q       = ws + 12416;    // 1024
  float* scores  = ws + 13440;    // 16*2048 = 32768
  float* o       = ws + 46208;    // 1024
  float* ff1     = ws + 47232;    // 4096

  float* logits       = (float*)d_out;
  float* h_out_base   = logits + VOCAB;
  float* conv_out_base= h_out_base + (size_t)N_LAYERS * D_INNER * D_STATE;

  k_embed<<<4, 256, 0, stream>>>(tok, emb, pos_scale, x);

  for (int l = 0; l < N_LAYERS; ++l) {
    // ---- Mamba block ----
    k_layernorm<<<1, 1024, 0, stream>>>(x, self_nw + l * D_MODEL, self_nb + l * D_MODEL, xn);
    k_gemv<0><<<512, 256, 0, stream>>>(in_proj_w + (size_t)l * 2 * D_INNER * D_MODEL,
                                       xn, nullptr, xz, 2 * D_INNER, D_MODEL);
    k_mamba_conv<<<8, 256, 0, stream>>>(xz, all_conv + (size_t)l * D_INNER * 3,
                                        conv_w + (size_t)l * D_INNER * D_CONV,
                                        conv_b + (size_t)l * D_INNER,
                                        x_conv, conv_out_base + (size_t)l * D_INNER * 3);
    k_gemv<0><<<12, 256, 0, stream>>>(x_proj_w + (size_t)l * (DT_RANK + 2 * D_STATE) * D_INNER,
                                      x_conv, nullptr, x_dbl, DT_RANK + 2 * D_STATE, D_INNER);
    k_gemv<3><<<256, 256, 0, stream>>>(dt_proj_w + (size_t)l * D_INNER * DT_RANK,
                                       x_dbl, dt_proj_b + (size_t)l * D_INNER, dt,
                                       D_INNER, DT_RANK);
    k_ssm<<<8, 256, 0, stream>>>(dt, x_dbl, A_log + (size_t)l * D_INNER * D_STATE,
                                 all_h + (size_t)l * D_INNER * D_STATE,
                                 Dp + (size_t)l * D_INNER, x_conv, xz,
                                 h_out_base + (size_t)l * D_INNER * D_STATE, y);
    k_gemv<1><<<128, 256, 0, stream>>>(mamba_ow + (size_t)l * D_MODEL * D_INNER,
                                       y, nullptr, x, D_MODEL, D_INNER);
    // ---- Cross-attention block ----
    k_layernorm<<<1, 1024, 0, stream>>>(x, cross_nw + l * D_MODEL, cross_nb + l * D_MODEL, xn);
    k_gemv<0><<<128, 256, 0, stream>>>(wq + (size_t)l * D_MODEL * D_MODEL,
                                       xn, bq + (size_t)l * D_MODEL, q, D_MODEL, D_MODEL);
    k_attn_scores<<<dim3(N_HEADS, SRC_LEN / 8), 256, 0, stream>>>(
        q, all_k + (size_t)l * N_HEADS * SRC_LEN * HEAD_DIM, scores);
    k_softmax<<<N_HEADS, 1024, 0, stream>>>(scores);
    k_attn_av<<<N_HEADS, 256, 0, stream>>>(
        scores, all_v + (size_t)l * N_HEADS * SRC_LEN * HEAD_DIM, o);
    k_gemv<1><<<128, 256, 0, stream>>>(attn_ow + (size_t)l * D_MODEL * D_MODEL,
                                       o, attn_ob + (size_t)l * D_MODEL, x, D_MODEL, D_MODEL);
    // ---- FFN block ----
    k_layernorm<<<1, 1024, 0, stream>>>(x, ff_nw + l * D_MODEL, ff_nb + l * D_MODEL, xn);
    k_gemv<2><<<512, 256, 0, stream>>>(ff_w1 + (size_t)l * D_FF * D_MODEL,
                                       xn, ff_b1 + (size_t)l * D_FF, ff1, D_FF, D_MODEL);
    k_gemv<1><<<512, 256, 0, stream>>>(ff_w2 + (size_t)l * D_MODEL * D_FF,
                                       ff1, ff_b2 + (size_t)l * D_MODEL, x, D_MODEL, D_FF);
  }

  // ---- final norm + vocab projection (WMMA) ----
  k_layernorm<<<1, 1024, 0, stream>>>(x, fn_w, fn_b, xn);
  int nwaves = (VOCAB + 15) / 16;             // 3142 waves
  int nblocks = (nwaves + 7) / 8;             // 8 waves / 256-thread block
  k_logits_wmma<<<nblocks, 256, 0, stream>>>(out_w, xn, out_b, logits, VOCAB);
}